// LIF_ODE_49185965474312
// MI455X (gfx1250) — compile-verified
//
#include <hip/hip_runtime.h>

// ---------------------------------------------------------------------------
// LIF ODE on MI455X (gfx1250).
//
// V_RESET == V_REST == 0  =>  the float32 trajectory after every reset is
// bit-identical to the trajectory from t=0, so the spike train is exactly
// periodic: spikes at t = p, 2p, 3p, ... where p is the first step with
// v_new >= 1.0f (p = 220 at I = 1.5), or no spikes at all if the float32
// iterate reaches a fixed point below threshold.
//
// Kernel 1 (1 lane): replay one inter-spike interval with the reference's
//   float32 arithmetic; compiler lowers this to the CDNA5 scalar-float SALU
//   (s_fmac_f32 chain) -> ~p dependent cycles, no VALU use.
// Kernel 2: zero-fill 14 MB via float4 (global_store_b128) -> pure store
//   stream, 14 MB / 23.3 TB/s ~= 0.6 us floor, zero arithmetic.
// Kernel 3: grid-stride scatter of 1.0f at the ~T/p spike positions
//   (~16 K dword stores at I=1.5); correct for any device-side p.
// ---------------------------------------------------------------------------

__global__ void lif_period_kernel(const float* __restrict__ I_ptr,
                                  int T,
                                  int* __restrict__ period_out) {
    if (blockIdx.x != 0 || threadIdx.x != 0) return;
    const float I     = I_ptr[0];
    const float alpha = 0.005f;            // DT/TAU = 1e-4 / 0.02
    float v = 0.0f;                        // V_REST
    int   p = 0;                           // 0 => never spikes
    for (int t = 1; t < T; ++t) {
        float vn = v + (-v + I) * alpha;   // same expression as reference
        if (vn >= 1.0f) { p = t; break; }  // spike -> reset to exactly 0
        if (vn == v)    { break; }         // float32 fixed point < threshold:
                                           // monotone iterate never spikes
        v = vn;
    }
    *period_out = p;
}

// Pure streaming zero-fill: one 128-bit store per lane, no VALU work.
__global__ void lif_zero_kernel(float* __restrict__ out, int T) {
    const int i0 = (blockIdx.x * blockDim.x + threadIdx.x) * 4;
    if (i0 + 3 < T) {
        *reinterpret_cast<float4*>(out + i0) = make_float4(0.f, 0.f, 0.f, 0.f);
    } else if (i0 < T) {
#pragma unroll
        for (int j = 0; j < 4; ++j)
            if (i0 + j < T) out[i0 + j] = 0.f;
    }
}

// Scatter 1.0f at t = (k+1)*p for all t < T. Grid-stride so the launch shape
// is independent of the device-computed period (graph-capture safe).
__global__ void lif_spike_scatter_kernel(float* __restrict__ out, int T,
                                         const int* __restrict__ period_ptr) {
    const int p = *period_ptr;             // uniform scalar load
    if (p <= 0) return;
    const long long stride =
        (long long)gridDim.x * blockDim.x * (long long)p;
    long long k = (long long)blockIdx.x * blockDim.x + threadIdx.x;
    for (long long t = (k + 1) * (long long)p; t < T; t += stride) {
        out[t] = 1.0f;
    }
}

extern "C" void kernel_launch(void* const* d_in, const int* in_sizes, int n_in,
                              void* d_out, int out_size, void* d_ws, size_t ws_size,
                              hipStream_t stream) {
    const float* d_I = (const float*)d_in[0];   // scalar input_current
    float* out       = (float*)d_out;
    int*   d_period  = (int*)d_ws;              // 4 bytes of scratch
    const int T      = out_size;                // == T_STEPS

    // Phase 1: serial exact replay of one inter-spike interval (SALU float).
    lif_period_kernel<<<1, 1, 0, stream>>>(d_I, T, d_period);

    // Phase 2: HBM-saturating zero stream, 16 B per lane, 8 waves per block.
    const int nvec  = (T + 3) / 4;
    const int block = 256;
    const int grid  = (nvec + block - 1) / block;
    lif_zero_kernel<<<grid, block, 0, stream>>>(out, T);

    // Phase 3: scatter the ~T/p ones (runs after phase 2 on the same stream).
    lif_spike_scatter_kernel<<<512, 256, 0, stream>>>(out, T, d_period);
}